// DRGNN_15341623181377
// MI455X (gfx1250) — compile-verified
//
#include <hip/hip_runtime.h>
#include <cstdint>

#define IN_DIM   128
#define H_DIM    128
#define OUT_DIM  40
#define MAX_ITER 50
#define TOL_F    1e-6f

typedef __attribute__((ext_vector_type(2))) float        v2f;
typedef __attribute__((ext_vector_type(8))) float        v8f;
typedef __attribute__((ext_vector_type(4))) unsigned int u32x4;
typedef __attribute__((ext_vector_type(8))) int          i32x8;
typedef __attribute__((ext_vector_type(4))) int          i32x4;

// Native fp32 WMMA: D(16x16,f32) = A(16x4,f32) * B(4x16,f32) + C
__device__ __forceinline__ v8f wmma_f32_k4(v2f a, v2f b, v8f c) {
    return __builtin_amdgcn_wmma_f32_16x16x4_f32(false, a, false, b, (short)0, c,
                                                 false, false);
}

// ---------------------------------------------------------------------------
// TDM: async-copy a [rows x 128] f32 tile (row stride 128) from global to LDS.
// D# packed per CDNA5 ISA 8.3/8.4 (Group0 128b, Group1 256b; groups 2/3 zero
// for a 2D tensor). Issue from ONE wave, then s_wait_tensorcnt + barrier.
// This toolchain exposes the 6-arg builtin:
//   (u32x4 g0, i32x8 g1, i32x4 g2, i32x4 g3, i32x8 extra, i32 cpol)
// ---------------------------------------------------------------------------
__device__ __forceinline__ void tdm_load_tile_f32(const float* gsrc,
                                                  unsigned int lds_off,
                                                  unsigned int rows) {
    const unsigned long long ga = (unsigned long long)(uintptr_t)gsrc;
    u32x4 g0;
    g0[0] = 1u;                                   // count=1, user descriptor
    g0[1] = lds_off;                              // lds_addr (bytes)
    g0[2] = (unsigned int)(ga & 0xFFFFFFFFu);     // global_addr[31:0]
    g0[3] = (unsigned int)((ga >> 32) & 0x1FFFFFFu) | (2u << 30);  // addr[56:32] | type=2
    i32x8 g1;
    g1[0] = (int)(2u << 16);                      // data_size=4B; mask=0; no pad/iterate
    g1[1] = (int)(128u << 16);                    // tensor_dim0[15:0]=128 (bits 63:48)
    g1[2] = (int)(rows << 16);                    // tensor_dim0 hi=0 | tensor_dim1 lo=rows
    g1[3] = (int)(128u << 16);                    // tensor_dim1 hi=0 | tile_dim0=128
    g1[4] = (int)(rows & 0xFFFFu);                // tile_dim1=rows, tile_dim2=0
    g1[5] = 128;                                  // tensor_dim0_stride = 128 elements
    g1[6] = 0;                                    // stride0 hi / dim1_stride lo
    g1[7] = 0;                                    // dim1_stride hi (unused for 2D)
    const i32x4 z4 = {0, 0, 0, 0};
    const i32x8 z8 = {0, 0, 0, 0, 0, 0, 0, 0};
    __builtin_amdgcn_tensor_load_to_lds(g0, g1, z4, z4, z8, 0);
}

// ---------------------------------------------------------------------------
// C[N x 128] = A[N x 128] * W^T (+ bvec), W row-major [128 x 128] (out,in).
// Block = 16 rows; 8 waves own one 16-col tile each; 32 fp32 WMMAs per tile.
// A tile staged into LDS by the Tensor Data Mover.
// ---------------------------------------------------------------------------
__global__ __launch_bounds__(256)
void gemm128_wmma(const float* __restrict__ A, const float* __restrict__ W,
                  const float* __restrict__ bvec, float* __restrict__ C) {
    __shared__ float As[16 * 128];
    const int row0 = blockIdx.x * 16;

    if (threadIdx.x < 32) {                       // one wave drives the TDM
        tdm_load_tile_f32(A + (size_t)row0 * 128,
                          (unsigned int)(uintptr_t)&As[0], 16u);
        __builtin_amdgcn_s_wait_tensorcnt(0);
    }
    __syncthreads();

    const int wv   = threadIdx.x >> 5;            // 0..7 -> column tile
    const int lane = threadIdx.x & 31;
    const int half = lane >> 4;                   // lanes 16-31 hold K+2,K+3
    const int mn   = lane & 15;                   // A: row M, B: col N
    const int col0 = wv * 16;
    const float* Wrow = W + (size_t)(col0 + mn) * 128;  // B[k][n] = W[n][k]

    v8f acc = {};
#pragma unroll
    for (int kk = 0; kk < 128; kk += 4) {
        const int k0 = kk + half * 2;
        v2f a = *(const v2f*)&As[mn * 128 + k0];
        v2f b = *(const v2f*)&Wrow[k0];
        acc = wmma_f32_k4(a, b, acc);
    }

    const float bv      = bvec ? bvec[col0 + mn] : 0.0f;
    const int   rowbase = row0 + half * 8;        // C: VGPR r -> rows r / r+8
#pragma unroll
    for (int r = 0; r < 8; ++r)
        C[(size_t)(rowbase + r) * 128 + col0 + mn] = acc[r] + bv;
}

// ---------------------------------------------------------------------------
// out[N x 40] = relu(U) * dec_W^T + dec_b.  3 waves/block; 3rd tile masked to
// OUT_DIM=40.  U tile staged raw via TDM; ReLU applied at fragment read.
// ---------------------------------------------------------------------------
__global__ __launch_bounds__(96)
void decoder_wmma(const float* __restrict__ U, const float* __restrict__ Wd,
                  const float* __restrict__ bd, float* __restrict__ out) {
    __shared__ float Us[16 * 128];
    const int row0 = blockIdx.x * 16;

    if (threadIdx.x < 32) {
        tdm_load_tile_f32(U + (size_t)row0 * 128,
                          (unsigned int)(uintptr_t)&Us[0], 16u);
        __builtin_amdgcn_s_wait_tensorcnt(0);
    }
    __syncthreads();

    const int wv   = threadIdx.x >> 5;            // 0..2
    const int lane = threadIdx.x & 31;
    const int half = lane >> 4;
    const int mn   = lane & 15;
    const int col  = wv * 16 + mn;
    const bool valid = (col < OUT_DIM);
    const float* Wrow = Wd + (size_t)(valid ? col : 0) * 128;

    v8f acc = {};
#pragma unroll
    for (int kk = 0; kk < 128; kk += 4) {
        const int k0 = kk + half * 2;
        v2f a = *(const v2f*)&Us[mn * 128 + k0];
        a[0] = fmaxf(a[0], 0.0f);                 // relu on the fly
        a[1] = fmaxf(a[1], 0.0f);
        v2f b = valid ? *(const v2f*)&Wrow[k0] : (v2f){0.0f, 0.0f};
        acc = wmma_f32_k4(a, b, acc);
    }

    if (valid) {
        const float bv      = bd[col];
        const int   rowbase = row0 + half * 8;
#pragma unroll
        for (int r = 0; r < 8; ++r)
            out[(size_t)(rowbase + r) * OUT_DIM + col] = acc[r] + bv;
    }
}

// ---------------------------------------------------------------------------
// Fixed-point loop kernels
// ---------------------------------------------------------------------------
__global__ void init_kernel(const float4* __restrict__ u0, float4* __restrict__ u,
                            unsigned* __restrict__ flags, int n4) {
    int i      = blockIdx.x * blockDim.x + threadIdx.x;
    int stride = gridDim.x * blockDim.x;
    for (; i < n4; i += stride) u[i] = u0[i];
    if (blockIdx.x == 0 && threadIdx.x == 0) { flags[0] = 0u; flags[1] = 0u; }
}

// initial prep (iteration 0 inputs): u_half = 2*relu(u) - u - bias ; agg = 0
__global__ void prep_kernel(const float4* __restrict__ u, const float4* __restrict__ bias,
                            float4* __restrict__ uhalf, float4* __restrict__ agg, int n4) {
    int i      = blockIdx.x * blockDim.x + threadIdx.x;
    int stride = gridDim.x * blockDim.x;
    for (; i < n4; i += stride) {
        float4 uv = u[i], bv = bias[i], r;
        r.x = 2.0f * fmaxf(uv.x, 0.0f) - uv.x - bv.x;
        r.y = 2.0f * fmaxf(uv.y, 0.0f) - uv.y - bv.y;
        r.z = 2.0f * fmaxf(uv.z, 0.0f) - uv.z - bv.z;
        r.w = 2.0f * fmaxf(uv.w, 0.0f) - uv.w - bv.w;
        uhalf[i] = r;
        agg[i]   = make_float4(0.0f, 0.0f, 0.0f, 0.0f);
    }
}

// one wave per edge; each lane owns 4 consecutive features (32*4 = 128)
__global__ __launch_bounds__(256)
void scatter_kernel(const int* __restrict__ src, const int* __restrict__ dst,
                    const float* __restrict__ ew, const float* __restrict__ uhalf,
                    float* __restrict__ agg, int E) {
    const int gtid   = blockIdx.x * blockDim.x + threadIdx.x;
    const int wave   = gtid >> 5;
    const int lane   = gtid & 31;
    const int nwaves = (gridDim.x * blockDim.x) >> 5;
    for (int e = wave; e < E; e += nwaves) {
        const int   s = src[e];
        const int   d = dst[e];
        const float w = ew[e];
        float4 v = ((const float4*)(uhalf + (size_t)s * 128))[lane];
        float* ap = agg + (size_t)d * 128 + lane * 4;
        __hip_atomic_fetch_add(ap + 0, w * v.x, __ATOMIC_RELAXED, __HIP_MEMORY_SCOPE_AGENT);
        __hip_atomic_fetch_add(ap + 1, w * v.y, __ATOMIC_RELAXED, __HIP_MEMORY_SCOPE_AGENT);
        __hip_atomic_fetch_add(ap + 2, w * v.z, __ATOMIC_RELAXED, __HIP_MEMORY_SCOPE_AGENT);
        __hip_atomic_fetch_add(ap + 3, w * v.w, __ATOMIC_RELAXED, __HIP_MEMORY_SCOPE_AGENT);
    }
}

// Fused update: un = 2*(u_half + c*agg)/gamma - 2*relu(u) + u ; err = max|u-un|;
// freeze with done flag from PREVIOUS iteration; then emit next iteration's
// u_half and zeroed agg in the same pass (saves a full elementwise kernel).
__global__ __launch_bounds__(256)
void update_kernel(float4* __restrict__ u, float4* __restrict__ uhalf,
                   float4* __restrict__ agg, const float4* __restrict__ bias,
                   const float* __restrict__ beta_p, const float* __restrict__ pg_p,
                   unsigned* __restrict__ flags, int n4) {
    const float c     = 2.0f / (1.0f + __expf(-beta_p[0])) - 1.0f;
    const float inv_g = 1.0f / (1.0f + fabsf(c) + 1.0f / (1.0f + __expf(-pg_p[0])));
    const unsigned done = flags[1];

    float lmax = 0.0f;
    int i      = blockIdx.x * blockDim.x + threadIdx.x;
    int stride = gridDim.x * blockDim.x;
    for (; i < n4; i += stride) {
        float4 uv = u[i], uh = uhalf[i], ag = agg[i], bv = bias[i], un, nx;
        un.x = 2.0f * (uh.x + c * ag.x) * inv_g - 2.0f * fmaxf(uv.x, 0.0f) + uv.x;
        un.y = 2.0f * (uh.y + c * ag.y) * inv_g - 2.0f * fmaxf(uv.y, 0.0f) + uv.y;
        un.z = 2.0f * (uh.z + c * ag.z) * inv_g - 2.0f * fmaxf(uv.z, 0.0f) + uv.z;
        un.w = 2.0f * (uh.w + c * ag.w) * inv_g - 2.0f * fmaxf(uv.w, 0.0f) + uv.w;
        lmax = fmaxf(lmax, fabsf(uv.x - un.x));
        lmax = fmaxf(lmax, fabsf(uv.y - un.y));
        lmax = fmaxf(lmax, fabsf(uv.z - un.z));
        lmax = fmaxf(lmax, fabsf(uv.w - un.w));
        float4 nu;
        nu.x = done ? uv.x : un.x;  nu.y = done ? uv.y : un.y;
        nu.z = done ? uv.z : un.z;  nu.w = done ? uv.w : un.w;
        u[i] = nu;
        nx.x = 2.0f * fmaxf(nu.x, 0.0f) - nu.x - bv.x;   // next iteration's u_half
        nx.y = 2.0f * fmaxf(nu.y, 0.0f) - nu.y - bv.y;
        nx.z = 2.0f * fmaxf(nu.z, 0.0f) - nu.z - bv.z;
        nx.w = 2.0f * fmaxf(nu.w, 0.0f) - nu.w - bv.w;
        uhalf[i] = nx;
        agg[i]   = make_float4(0.0f, 0.0f, 0.0f, 0.0f);
    }

    __shared__ float red[256];
    red[threadIdx.x] = lmax;
    __syncthreads();
    for (int s = 128; s > 0; s >>= 1) {
        if (threadIdx.x < s) red[threadIdx.x] = fmaxf(red[threadIdx.x], red[threadIdx.x + s]);
        __syncthreads();
    }
    if (threadIdx.x == 0)
        atomicMax(&flags[0], __float_as_uint(red[0]));   // non-negative: bit order == value order
}

__global__ void finish_kernel(unsigned* __restrict__ flags) {
    if (__uint_as_float(flags[0]) <= TOL_F) flags[1] = 1u;
    flags[0] = 0u;
}

// ---------------------------------------------------------------------------
extern "C" void kernel_launch(void* const* d_in, const int* in_sizes, int n_in,
                              void* d_out, int out_size, void* d_ws, size_t ws_size,
                              hipStream_t stream) {
    const float* x     = (const float*)d_in[0];
    const int*   eidx  = (const int*)d_in[1];   // [2,E] flat: src then dst
    const float* ew    = (const float*)d_in[2];
    const float* u0    = (const float*)d_in[3];
    const float* encW  = (const float*)d_in[4];
    const float* encB  = (const float*)d_in[5];
    const float* biasW = (const float*)d_in[6];
    const float* decW  = (const float*)d_in[7];
    const float* decB  = (const float*)d_in[8];
    const float* beta  = (const float*)d_in[9];
    const float* pg    = (const float*)d_in[10];

    const int    N  = in_sizes[0] / IN_DIM;   // 50000
    const int    E  = in_sizes[2];            // 800000
    const size_t NH = (size_t)N * H_DIM;

    float* ws    = (float*)d_ws;
    float* bias  = ws;                 // [N,H]
    float* u     = ws + NH;            // [N,H]
    float* uhalf = ws + 2 * NH;        // [N,H]  (also holds h = enc(x))
    float* agg   = ws + 3 * NH;        // [N,H]
    unsigned* flags = (unsigned*)(ws + 4 * NH);  // [0]=err bits, [1]=done

    const int n4        = (int)(NH / 4);
    const int eltBlocks = 4096;
    const int rowTiles  = N / 16;      // 3125

    init_kernel<<<eltBlocks, 256, 0, stream>>>((const float4*)u0, (float4*)u, flags, n4);

    // h = x @ enc_W^T + enc_b   (h lives in uhalf slot until the loop starts)
    gemm128_wmma<<<rowTiles, 256, 0, stream>>>(x, encW, encB, uhalf);
    // bias = h @ bias_W^T
    gemm128_wmma<<<rowTiles, 256, 0, stream>>>(uhalf, biasW, nullptr, bias);

    // iteration-0 inputs
    prep_kernel<<<eltBlocks, 256, 0, stream>>>((const float4*)u, (const float4*)bias,
                                               (float4*)uhalf, (float4*)agg, n4);

    for (int it = 0; it < MAX_ITER; ++it) {
        scatter_kernel<<<2048, 256, 0, stream>>>(eidx, eidx + E, ew, uhalf, agg, E);
        update_kernel<<<eltBlocks, 256, 0, stream>>>((float4*)u, (float4*)uhalf,
                                                     (float4*)agg, (const float4*)bias,
                                                     beta, pg, flags, n4);
        finish_kernel<<<1, 1, 0, stream>>>(flags);
    }

    decoder_wmma<<<rowTiles, 96, 0, stream>>>(u, decW, decB, (float*)d_out);
}